// GraphAggregator_layer_3942779978055
// MI455X (gfx1250) — compile-verified
//
#include <hip/hip_runtime.h>

typedef float v2f __attribute__((ext_vector_type(2)));
typedef float v4f __attribute__((ext_vector_type(4)));
typedef float v8f __attribute__((ext_vector_type(8)));
typedef int   v4i __attribute__((ext_vector_type(4)));

#define D 128
#define G 8192
#define TILES 16
#define ROWS (TILES * 16)   // 256 rows per block
#define LSTR (D + 4)        // padded LDS row stride -> conflict-free b64 reads

// ---- CDNA5 async global->LDS path (probe: guarded, with sync fallback) ----
#if defined(__has_builtin)
#if __has_builtin(__builtin_amdgcn_global_load_async_to_lds_b128)
#define HAVE_ASYNC_LDS 1
#endif
#endif
#ifndef HAVE_ASYNC_LDS
#define HAVE_ASYNC_LDS 0
#endif

#define AS1 __attribute__((address_space(1)))
#define AS3 __attribute__((address_space(3)))

__device__ __forceinline__ void async_stage_wait() {
#if HAVE_ASYNC_LDS
#if defined(__has_builtin) && __has_builtin(__builtin_amdgcn_s_wait_asynccnt)
  __builtin_amdgcn_s_wait_asynccnt(0);
#else
  asm volatile("s_wait_asynccnt 0x0" ::: "memory");
#endif
#endif
}

// Stage one 16x128 fp32 tile into (padded) LDS. 256 threads x 2 b128 slots.
// GUARD: mask rows >= N (unmasked rows may hold garbage in async mode; those
// rows are excluded later in the pooling pass, and garbage in A row m only
// pollutes output row m, so this is safe).
template <bool GUARD>
__device__ __forceinline__ void stage_tile(const float* __restrict__ src,
                                           long long row0, int N,
                                           float* __restrict__ buf, int tid) {
#pragma unroll
  for (int i = 0; i < 2; ++i) {
    const int s = tid + i * 256;  // 0..511 float4 slots (32 per row)
    const int r = s >> 5;
    const long long grow = row0 + r;
#if HAVE_ASYNC_LDS
    if (!GUARD || grow < N) {
      AS1 v4i* g = (AS1 v4i*)(src + grow * (long long)D) + (s & 31);
      AS3 v4i* l = (AS3 v4i*)(buf + r * LSTR) + (s & 31);
      __builtin_amdgcn_global_load_async_to_lds_b128(g, l, 0, 0);
    }
#else
    v4f v = {0.f, 0.f, 0.f, 0.f};
    if (!GUARD || grow < N) v = ((const v4f*)(src + grow * (long long)D))[s & 31];
    ((v4f*)(buf + r * LSTR))[s & 31] = v;
#endif
  }
}

__device__ __forceinline__ void pooled_fadd(float* p, float v) {
  __hip_atomic_fetch_add(p, v, __ATOMIC_RELAXED, __HIP_MEMORY_SCOPE_AGENT);
}

// ---------------------------------------------------------------------------
// Kernel 0: zero the pooled accumulator [G, D] in workspace
// ---------------------------------------------------------------------------
__global__ __launch_bounds__(256) void zero_pooled(float* __restrict__ pooled) {
  v4f* p = (v4f*)pooled;
  int base = blockIdx.x * blockDim.x + threadIdx.x;
  v4f z = {0.f, 0.f, 0.f, 0.f};
#pragma unroll
  for (int i = 0; i < 4; ++i) p[base + i * 65536] = z;
}

// ---------------------------------------------------------------------------
// Kernel 1: fused  out = (H@W + b) * relu(H@Wg + bg)  + sorted-segment pooling
// Block = 8 waves, 256 rows. Wave w owns output cols [16w,16w+16) for BOTH
// weight matrices; B fragments live in registers. A tiles double-buffered with
// async global->LDS loads overlapped against WMMA + pooling.
// ---------------------------------------------------------------------------
__global__ __launch_bounds__(256)
void fused_node_pool(const float* __restrict__ H, const int* __restrict__ seg,
                     const float* __restrict__ W, const float* __restrict__ b,
                     const float* __restrict__ Wg, const float* __restrict__ bg,
                     float* __restrict__ pooled, int N) {
  __shared__ float Abuf[2][16 * LSTR];  // double-buffered A tiles
  __shared__ float Olds[16 * LSTR];     // gated output tile
  __shared__ int SegLds[ROWS];          // chunk's segment ids

  const int tid = threadIdx.x;
  const int wave = tid >> 5;
  const int lane = tid & 31;
  const int lhalf = lane >> 4;  // K-half for A/B frags, M-half for C
  const int l15 = lane & 15;
  const int col = wave * 16 + l15;
  const long long chunk0 = (long long)blockIdx.x * ROWS;

  // stage segment ids for the whole chunk (coalesced, once)
  SegLds[tid] = (chunk0 + tid < N) ? seg[chunk0 + tid] : -1;

  // --- preload B fragments for W and W_gate (ISA 7.12.2 B layout) ---
  v2f bw[32], bgt[32];
#pragma unroll
  for (int k = 0; k < 32; ++k) {
    const int kr = 4 * k + 2 * lhalf;
    bw[k].x = W[(kr + 0) * D + col];
    bw[k].y = W[(kr + 1) * D + col];
    bgt[k].x = Wg[(kr + 0) * D + col];
    bgt[k].y = Wg[(kr + 1) * D + col];
  }
  const float bias_s = b[col];
  const float bias_g = bg[col];

  // pooling state: threads 0..127 each own one output column
  float racc = 0.f;
  int cur = -1;
  const int pc = tid & 127;

  // prologue: stage tile 0
  stage_tile<true>(H, chunk0, N, Abuf[0], tid);
  async_stage_wait();
  __syncthreads();

  for (int t = 0; t < TILES; ++t) {
    const long long row0 = chunk0 + (long long)t * 16;

    // issue next tile's staging (overlaps with WMMA + pooling below)
    if (t + 1 < TILES)
      stage_tile<true>(H, row0 + 16, N, Abuf[(t + 1) & 1], tid);

    const float* A = Abuf[t & 1];

    // --- 32 K-steps of fp32 WMMA, dual accumulators (state + gate) ---
    v8f accs = {};
    v8f accg = {};
#pragma unroll
    for (int k = 0; k < 32; ++k) {
      const v2f a = *(const v2f*)(&A[l15 * LSTR + 4 * k + 2 * lhalf]);
      accs = __builtin_amdgcn_wmma_f32_16x16x4_f32(false, a, false, bw[k],
                                                   (short)0, accs, false, false);
      accg = __builtin_amdgcn_wmma_f32_16x16x4_f32(false, a, false, bgt[k],
                                                   (short)0, accg, false, false);
    }

    // --- epilogue: bias + relu gate -> LDS tile for pooling ---
#pragma unroll
    for (int j = 0; j < 8; ++j) {
      const float s = accs[j] + bias_s;
      float g = accg[j] + bias_g;
      g = g > 0.f ? g : 0.f;
      const int m = j + 8 * lhalf;  // row within tile (C/D layout)
      Olds[m * LSTR + col] = s * g;
    }
    __syncthreads();

    // --- sorted-segment pooling: run-length accumulate, flush on change ---
    if (tid < 128) {
      for (int r = 0; r < 16; ++r) {
        if (row0 + r < N) {
          const int sg = SegLds[t * 16 + r];
          if (sg != cur) {
            if (cur >= 0) pooled_fadd(&pooled[(long long)cur * D + pc], racc);
            racc = 0.f;
            cur = sg;
          }
          racc += Olds[r * LSTR + pc];
        }
      }
    }

    // next tile's async stage must have landed before anyone reads it;
    // barrier also releases Olds for overwrite
    async_stage_wait();
    __syncthreads();
  }
  if (tid < 128 && cur >= 0) pooled_fadd(&pooled[(long long)cur * D + pc], racc);
}

// ---------------------------------------------------------------------------
// Kernel 2: out = pooled @ W_final + b_final   [8192,128]x[128,128]
// ---------------------------------------------------------------------------
__global__ __launch_bounds__(256)
void final_gemm(const float* __restrict__ P, const float* __restrict__ Wf,
                const float* __restrict__ bf, float* __restrict__ out) {
  __shared__ float Abuf[2][16 * LSTR];

  const int tid = threadIdx.x;
  const int wave = tid >> 5;
  const int lane = tid & 31;
  const int lhalf = lane >> 4;
  const int l15 = lane & 15;
  const int col = wave * 16 + l15;
  const long long chunk0 = (long long)blockIdx.x * ROWS;

  v2f bwf[32];
#pragma unroll
  for (int k = 0; k < 32; ++k) {
    const int kr = 4 * k + 2 * lhalf;
    bwf[k].x = Wf[(kr + 0) * D + col];
    bwf[k].y = Wf[(kr + 1) * D + col];
  }
  const float bias = bf[col];

  stage_tile<false>(P, chunk0, 0x7fffffff, Abuf[0], tid);
  async_stage_wait();
  __syncthreads();

  for (int t = 0; t < TILES; ++t) {
    const long long row0 = chunk0 + (long long)t * 16;

    if (t + 1 < TILES)
      stage_tile<false>(P, row0 + 16, 0x7fffffff, Abuf[(t + 1) & 1], tid);

    const float* A = Abuf[t & 1];

    v8f acc = {};
#pragma unroll
    for (int k = 0; k < 32; ++k) {
      const v2f a = *(const v2f*)(&A[l15 * LSTR + 4 * k + 2 * lhalf]);
      acc = __builtin_amdgcn_wmma_f32_16x16x4_f32(false, a, false, bwf[k],
                                                  (short)0, acc, false, false);
    }

#pragma unroll
    for (int j = 0; j < 8; ++j) {
      const int m = j + 8 * lhalf;
      out[(row0 + m) * D + col] = acc[j] + bias;
    }

    async_stage_wait();
    __syncthreads();
  }
}

// ---------------------------------------------------------------------------
extern "C" void kernel_launch(void* const* d_in, const int* in_sizes, int n_in,
                              void* d_out, int out_size, void* d_ws, size_t ws_size,
                              hipStream_t stream) {
  const float* H  = (const float*)d_in[0];
  const int* seg  = (const int*)d_in[1];
  // d_in[2] = num_graphs scalar (compile-time constant G = 8192)
  const float* W  = (const float*)d_in[3];
  const float* b  = (const float*)d_in[4];
  const float* Wg = (const float*)d_in[5];
  const float* bg = (const float*)d_in[6];
  const float* Wf = (const float*)d_in[7];
  const float* bf = (const float*)d_in[8];
  float* out = (float*)d_out;
  float* pooled = (float*)d_ws;  // G*D floats = 4 MB scratch

  const int N = in_sizes[0] / D;

  zero_pooled<<<256, 256, 0, stream>>>(pooled);

  const int nb = (N + ROWS - 1) / ROWS;
  fused_node_pool<<<nb, 256, 0, stream>>>(H, seg, W, b, Wg, bg, pooled, N);

  final_gemm<<<G / ROWS, 256, 0, stream>>>(pooled, Wf, bf, out);
}